// MultiScaleAttention_31714038513666
// MI455X (gfx1250) — compile-verified
//
#include <hip/hip_runtime.h>
#include <hip/hip_bf16.h>
#include <math.h>

typedef __attribute__((ext_vector_type(2))) float v2f;
typedef __attribute__((ext_vector_type(8))) float v8f;

// ---- problem constants (match reference) ----
constexpr int BB     = 16;
constexpr int T0C    = 8, H0C = 14, W0C = 14;
constexpr int DIM    = 768;
constexpr int HEADS  = 8;
constexpr int HD     = 96;              // DIM / HEADS
constexpr int NTOK   = T0C*H0C*W0C + 1; // 1569
constexpr int LQ     = NTOK;            // q pooled (stride 1,1,1) + cls
constexpr int KTD    = 8, KHD = 7, KWD = 7;
constexpr int LK     = KTD*KHD*KWD + 1; // 393
constexpr int BH     = BB*HEADS;        // 128
constexpr int QBODY  = NTOK - 1;        // 1568
constexpr int KBODY  = LK - 1;          // 392
constexpr int QKVC   = 3*DIM;           // 2304
constexpr int MROWS  = BB*NTOK;         // 25104
constexpr float SCALE = 0.10206207261596577f;   // 96^-0.5

__device__ inline v8f wmma_f32(v2f a, v2f b, v8f c) {
  // V_WMMA_F32_16X16X4_F32 : D = A(16x4) * B(4x16) + C(16x16), all f32
  return __builtin_amdgcn_wmma_f32_16x16x4_f32(
      false, a, false, b, (short)0, c, false, false);
}

// ---------------------------------------------------------------------------
// K1: qkv = x @ qkv_w + qkv_b  (M=25104, K=768, N=2304), 64x64 tile per wave
// ---------------------------------------------------------------------------
__global__ void __launch_bounds__(32, 1)
gemm_qkv_kernel(const float* __restrict__ X,
                const float* __restrict__ W,
                const float* __restrict__ bias,
                float* __restrict__ C) {
  const int lane = threadIdx.x;
  const int half = lane >> 4;
  const int l15  = lane & 15;
  const int m0 = blockIdx.x * 64;
  const int n0 = blockIdx.y * 64;
  v8f acc[4][4] = {};
  int rowA[4];
#pragma unroll
  for (int i = 0; i < 4; ++i) rowA[i] = min(m0 + i*16 + l15, MROWS-1);
#pragma unroll 1
  for (int kk = 0; kk < DIM; kk += 4) {
    const int ka = kk + half*2;
    v2f a[4], b[4];
#pragma unroll
    for (int i = 0; i < 4; ++i) {
      a[i].x = X[(size_t)rowA[i]*DIM + ka];
      a[i].y = X[(size_t)rowA[i]*DIM + ka + 1];
    }
#pragma unroll
    for (int j = 0; j < 4; ++j) {
      const int cb = n0 + j*16 + l15;
      b[j].x = W[(size_t)ka*QKVC + cb];
      b[j].y = W[(size_t)(ka+1)*QKVC + cb];
    }
#pragma unroll
    for (int i = 0; i < 4; ++i)
#pragma unroll
      for (int j = 0; j < 4; ++j) acc[i][j] = wmma_f32(a[i], b[j], acc[i][j]);
  }
#pragma unroll
  for (int i = 0; i < 4; ++i) {
#pragma unroll
    for (int ii = 0; ii < 8; ++ii) {
      const int r = m0 + i*16 + ii + half*8;
      if (r < MROWS) {
#pragma unroll
        for (int j = 0; j < 4; ++j) {
          const int cb = n0 + j*16 + l15;
          C[(size_t)r*QKVC + cb] = acc[i][j][ii] + bias[cb];
        }
      }
    }
  }
}

// ---------------------------------------------------------------------------
// K2: depthwise 3x3x3 pool + LayerNorm over HD channels (q / k / v)
// ---------------------------------------------------------------------------
__global__ void pool_ln_kernel(const float* __restrict__ qkv,
                               const float* __restrict__ w27,
                               const float* __restrict__ gamma,
                               const float* __restrict__ beta,
                               float* __restrict__ out,
                               int which, int sT, int sH, int sW,
                               int nT, int nH, int nW, int Lout) {
  const int bh  = blockIdx.y;
  const int b   = bh / HEADS, h = bh % HEADS;
  const int tok = blockIdx.x;
  const int c   = threadIdx.x;
  const float* base = qkv + (size_t)b*NTOK*QKVC + which*DIM + h*HD;
  float val;
  if (tok == 0) {
    val = base[c];                        // cls token passes through
  } else {
    const int idx = tok - 1;
    const int ot  = idx / (nH*nW);
    const int rem = idx % (nH*nW);
    const int oy  = rem / nW, ox = rem % nW;
    float s = 0.f;
    for (int dz = 0; dz < 3; ++dz) {
      const int it = ot*sT - 1 + dz;
      if (it < 0 || it >= T0C) continue;
      for (int dy = 0; dy < 3; ++dy) {
        const int iy = oy*sH - 1 + dy;
        if (iy < 0 || iy >= H0C) continue;
        for (int dx = 0; dx < 3; ++dx) {
          const int ix = ox*sW - 1 + dx;
          if (ix < 0 || ix >= W0C) continue;
          const int n = 1 + (it*H0C + iy)*W0C + ix;
          s += w27[c*27 + (dz*3 + dy)*3 + dx] * base[(size_t)n*QKVC + c];
        }
      }
    }
    val = s;
  }
  __shared__ float sh[HD];
  __shared__ float mv[2];
  sh[c] = val;
  __syncthreads();
  if (c == 0) {
    float m = 0.f;
    for (int i = 0; i < HD; ++i) m += sh[i];
    m *= (1.0f/HD);
    float v = 0.f;
    for (int i = 0; i < HD; ++i) { const float d = sh[i]-m; v += d*d; }
    v *= (1.0f/HD);
    mv[0] = m; mv[1] = rsqrtf(v + 1e-5f);
  }
  __syncthreads();
  out[((size_t)bh*Lout + tok)*HD + c] = (val - mv[0])*mv[1]*gamma[c] + beta[c];
}

// ---------------------------------------------------------------------------
// K3: rel-pos dot products.  rel layout [BH][QBODY][22]:
//     [0..6]=rel_h(ky), [7..13]=rel_w(kx), [14..21]=rel_t(kt)
// ---------------------------------------------------------------------------
__global__ void rel_compute_kernel(const float* __restrict__ qp,
                                   const float* __restrict__ rph,
                                   const float* __restrict__ rpw,
                                   const float* __restrict__ rpt,
                                   float* __restrict__ rel) {
  const long long total = (long long)BH*QBODY*22;
  const long long gid = (long long)blockIdx.x*blockDim.x + threadIdx.x;
  if (gid >= total) return;
  const int e    = (int)(gid % 22);
  const long long r1 = gid / 22;
  const int qb   = (int)(r1 % QBODY);
  const int bh   = (int)(r1 / QBODY);
  const int t = qb / (H0C*W0C);
  const int r2 = qb % (H0C*W0C);
  const int y = r2 / W0C, x = r2 % W0C;
  const float* q = qp + ((size_t)bh*LQ + 1 + qb)*HD;
  const float* R;
  if (e < 7)       R = rph + (size_t)(y - 2*e       + 12)*HD;   // _rel_idx(14,7)
  else if (e < 14) R = rpw + (size_t)(x - 2*(e-7)   + 12)*HD;
  else             R = rpt + (size_t)(t -   (e-14)  +  7)*HD;   // _rel_idx(8,8)
  float s = 0.f;
  for (int c = 0; c < HD; ++c) s += q[c]*R[c];
  rel[gid] = s;
}

// ---------------------------------------------------------------------------
// K4: attn = scale * q k^T  (+ fused rel-pos bias on body block)
//     M=1569, N=393, K=96; 64x64 tile per wave
// ---------------------------------------------------------------------------
__global__ void __launch_bounds__(32, 1)
attn_scores_kernel(const float* __restrict__ qp,
                   const float* __restrict__ kp,
                   const float* __restrict__ rel,
                   float* __restrict__ attn) {
  const int bh = blockIdx.z;
  const float* Q  = qp + (size_t)bh*LQ*HD;
  const float* Km = kp + (size_t)bh*LK*HD;
  float* A = attn + (size_t)bh*LQ*LK;
  const int lane = threadIdx.x;
  const int half = lane >> 4;
  const int l15  = lane & 15;
  const int m0 = blockIdx.x * 64;
  const int n0 = blockIdx.y * 64;
  v8f acc[4][4] = {};
  int rowA[4], colC[4];
#pragma unroll
  for (int i = 0; i < 4; ++i) rowA[i] = min(m0 + i*16 + l15, LQ-1);
#pragma unroll
  for (int j = 0; j < 4; ++j) colC[j] = min(n0 + j*16 + l15, LK-1);
#pragma unroll 1
  for (int kk = 0; kk < HD; kk += 4) {
    const int ka = kk + half*2;
    v2f a[4], b[4];
#pragma unroll
    for (int i = 0; i < 4; ++i) {
      a[i].x = Q[(size_t)rowA[i]*HD + ka];
      a[i].y = Q[(size_t)rowA[i]*HD + ka + 1];
    }
#pragma unroll
    for (int j = 0; j < 4; ++j) {
      b[j].x = Km[(size_t)colC[j]*HD + ka];       // B[k][n] = K[n][k]
      b[j].y = Km[(size_t)colC[j]*HD + ka + 1];
    }
#pragma unroll
    for (int i = 0; i < 4; ++i)
#pragma unroll
      for (int j = 0; j < 4; ++j) acc[i][j] = wmma_f32(a[i], b[j], acc[i][j]);
  }
#pragma unroll
  for (int i = 0; i < 4; ++i) {
#pragma unroll
    for (int ii = 0; ii < 8; ++ii) {
      const int r = m0 + i*16 + ii + half*8;
      if (r < LQ) {
        const float* rl = rel + ((size_t)bh*QBODY + (r-1))*22;  // valid iff r>=1
#pragma unroll
        for (int j = 0; j < 4; ++j) {
          const int cn = n0 + j*16 + l15;
          if (cn < LK) {
            float v = acc[i][j][ii] * SCALE;
            if (r >= 1 && cn >= 1) {
              const int kb = cn - 1;
              const int kt = kb / (KHD*KWD);
              const int rr = kb % (KHD*KWD);
              v += rl[rr / KWD] + rl[7 + rr % KWD] + rl[14 + kt];
            }
            A[(size_t)r*LK + cn] = v;
          }
        }
      }
    }
  }
}

// ---------------------------------------------------------------------------
// K5: row softmax over LK=393 — one wave per row
// ---------------------------------------------------------------------------
__global__ void softmax_kernel(float* __restrict__ attn) {
  const int bh  = blockIdx.y;
  const int row = blockIdx.x;
  float* a = attn + ((size_t)bh*LQ + row)*LK;
  const int lane = threadIdx.x;
  float m = -3.0e38f;
  for (int j = lane; j < LK; j += 32) m = fmaxf(m, a[j]);
  for (int off = 16; off > 0; off >>= 1) m = fmaxf(m, __shfl_xor(m, off, 32));
  float s = 0.f;
  for (int j = lane; j < LK; j += 32) { const float e = __expf(a[j]-m); a[j] = e; s += e; }
  for (int off = 16; off > 0; off >>= 1) s += __shfl_xor(s, off, 32);
  const float inv = 1.f/s;
  for (int j = lane; j < LK; j += 32) a[j] *= inv;
}

// ---------------------------------------------------------------------------
// K6: outh = attn @ v (+ residual q for body rows)
//     M=1569, N=96, K=393; 64x32 tile per wave
// ---------------------------------------------------------------------------
__global__ void __launch_bounds__(32, 1)
attn_v_kernel(const float* __restrict__ attn,
              const float* __restrict__ vp,
              const float* __restrict__ qp,
              float* __restrict__ outh) {
  const int bh = blockIdx.z;
  const float* A = attn + (size_t)bh*LQ*LK;
  const float* V = vp   + (size_t)bh*LK*HD;
  const float* Q = qp   + (size_t)bh*LQ*HD;
  float* O = outh + (size_t)bh*LQ*HD;
  const int lane = threadIdx.x;
  const int half = lane >> 4;
  const int l15  = lane & 15;
  const int m0 = blockIdx.x * 64;
  const int n0 = blockIdx.y * 32;
  v8f acc[4][2] = {};
  int rowA[4];
#pragma unroll
  for (int i = 0; i < 4; ++i) rowA[i] = min(m0 + i*16 + l15, LQ-1);
#pragma unroll 1
  for (int kk = 0; kk < 396; kk += 4) {            // K=393 padded to 396
    const int ka = kk + half*2;
    v2f a[4], b[2];
#pragma unroll
    for (int i = 0; i < 4; ++i) {
      a[i].x = (ka   < LK) ? A[(size_t)rowA[i]*LK + ka]     : 0.f;
      a[i].y = (ka+1 < LK) ? A[(size_t)rowA[i]*LK + ka + 1] : 0.f;
    }
#pragma unroll
    for (int j = 0; j < 2; ++j) {
      const int cb = n0 + j*16 + l15;
      b[j].x = (ka   < LK) ? V[(size_t)ka*HD + cb]     : 0.f;
      b[j].y = (ka+1 < LK) ? V[(size_t)(ka+1)*HD + cb] : 0.f;
    }
#pragma unroll
    for (int i = 0; i < 4; ++i)
#pragma unroll
      for (int j = 0; j < 2; ++j) acc[i][j] = wmma_f32(a[i], b[j], acc[i][j]);
  }
#pragma unroll
  for (int i = 0; i < 4; ++i) {
#pragma unroll
    for (int ii = 0; ii < 8; ++ii) {
      const int r = m0 + i*16 + ii + half*8;
      if (r < LQ) {
#pragma unroll
        for (int j = 0; j < 2; ++j) {
          const int cb = n0 + j*16 + l15;
          const float res = (r >= 1) ? Q[(size_t)r*HD + cb] : 0.f;
          O[(size_t)r*HD + cb] = acc[i][j][ii] + res;
        }
      }
    }
  }
}

// ---------------------------------------------------------------------------
// K7: out = concat_heads(outh) @ proj_w + proj_b  (M=25104, K=768, N=768)
//     64x64 tile per wave; A gathered from outh[b][h][n][c], k = h*96 + c
// ---------------------------------------------------------------------------
__global__ void __launch_bounds__(32, 1)
gemm_proj_kernel(const float* __restrict__ outh,
                 const float* __restrict__ W,
                 const float* __restrict__ bias,
                 float* __restrict__ Y) {
  const int lane = threadIdx.x;
  const int half = lane >> 4;
  const int l15  = lane & 15;
  const int m0 = blockIdx.x * 64;
  const int n0 = blockIdx.y * 64;
  v8f acc[4][4] = {};
  size_t rbase[4];
  int rowA[4];
#pragma unroll
  for (int i = 0; i < 4; ++i) {
    rowA[i] = min(m0 + i*16 + l15, MROWS-1);
    const int b_ = rowA[i] / NTOK, n_ = rowA[i] % NTOK;
    rbase[i] = (size_t)b_*HEADS*LQ*HD + (size_t)n_*HD;   // + h*LQ*HD + c
  }
#pragma unroll 1
  for (int kk = 0; kk < DIM; kk += 4) {
    const int ka = kk + half*2;
    const int h0 = ka / HD,     c0 = ka % HD;
    const int h1 = (ka+1) / HD, c1 = (ka+1) % HD;
    v2f a[4], b[4];
#pragma unroll
    for (int i = 0; i < 4; ++i) {
      a[i].x = outh[rbase[i] + (size_t)h0*LQ*HD + c0];
      a[i].y = outh[rbase[i] + (size_t)h1*LQ*HD + c1];
    }
#pragma unroll
    for (int j = 0; j < 4; ++j) {
      const int cb = n0 + j*16 + l15;
      b[j].x = W[(size_t)ka*DIM + cb];
      b[j].y = W[(size_t)(ka+1)*DIM + cb];
    }
#pragma unroll
    for (int i = 0; i < 4; ++i)
#pragma unroll
      for (int j = 0; j < 4; ++j) acc[i][j] = wmma_f32(a[i], b[j], acc[i][j]);
  }
#pragma unroll
  for (int i = 0; i < 4; ++i) {
#pragma unroll
    for (int ii = 0; ii < 8; ++ii) {
      const int r = m0 + i*16 + ii + half*8;
      if (r < MROWS) {
#pragma unroll
        for (int j = 0; j < 4; ++j) {
          const int cb = n0 + j*16 + l15;
          Y[(size_t)r*DIM + cb] = acc[i][j][ii] + bias[cb];
        }
      }
    }
  }
}

// ---------------------------------------------------------------------------
extern "C" void kernel_launch(void* const* d_in, const int* in_sizes, int n_in,
                              void* d_out, int out_size, void* d_ws, size_t ws_size,
                              hipStream_t stream) {
  const float* x     = (const float*)d_in[0];
  const float* qkv_w = (const float*)d_in[1];
  const float* qkv_b = (const float*)d_in[2];
  const float* pqw   = (const float*)d_in[3];
  const float* pkw   = (const float*)d_in[4];
  const float* pvw   = (const float*)d_in[5];
  const float* lnqg  = (const float*)d_in[6];
  const float* lnqb  = (const float*)d_in[7];
  const float* lnkg  = (const float*)d_in[8];
  const float* lnkb  = (const float*)d_in[9];
  const float* lnvg  = (const float*)d_in[10];
  const float* lnvb  = (const float*)d_in[11];
  const float* rph   = (const float*)d_in[12];
  const float* rpw   = (const float*)d_in[13];
  const float* rpt   = (const float*)d_in[14];
  const float* pjw   = (const float*)d_in[15];
  const float* pjb   = (const float*)d_in[16];
  float* out = (float*)d_out;
  float* ws  = (float*)d_ws;

  // workspace layout (floats); attn overlaps the dead qkv buffer
  const size_t region0 = (size_t)BH*LQ*LK;        // 78,926,976 >= qkv's 57,839,616
  float* qkv  = ws;
  float* attn = ws;
  float* qp   = ws + region0;
  float* kp   = qp + (size_t)BH*LQ*HD;
  float* vp   = kp + (size_t)BH*LK*HD;
  float* rel  = vp + (size_t)BH*LK*HD;
  float* outh = rel + (size_t)BH*QBODY*22;

  // 1) qkv projection (64x64 tiles)
  gemm_qkv_kernel<<<dim3((MROWS+63)/64, QKVC/64), 32, 0, stream>>>(x, qkv_w, qkv_b, qkv);

  // 2) attention pooling + LN (q stride 1; k/v stride (1,2,2))
  pool_ln_kernel<<<dim3(LQ, BH), HD, 0, stream>>>(qkv, pqw, lnqg, lnqb, qp, 0, 1,1,1, T0C,H0C,W0C, LQ);
  pool_ln_kernel<<<dim3(LK, BH), HD, 0, stream>>>(qkv, pkw, lnkg, lnkb, kp, 1, 1,2,2, KTD,KHD,KWD, LK);
  pool_ln_kernel<<<dim3(LK, BH), HD, 0, stream>>>(qkv, pvw, lnvg, lnvb, vp, 2, 1,2,2, KTD,KHD,KWD, LK);

  // 3) rel-pos dot products (needed by fused scores kernel)
  {
    const long long total = (long long)BH*QBODY*22;
    rel_compute_kernel<<<(unsigned)((total+255)/256), 256, 0, stream>>>(qp, rph, rpw, rpt, rel);
  }

  // 4) scores = scale * q k^T + rel bias  (qkv buffer dead; attn reuses it)
  attn_scores_kernel<<<dim3((LQ+63)/64, (LK+63)/64, BH), 32, 0, stream>>>(qp, kp, rel, attn);

  // 5) softmax rows
  softmax_kernel<<<dim3(LQ, BH), 32, 0, stream>>>(attn);

  // 6) attn @ v + residual (64x32 tiles)
  attn_v_kernel<<<dim3((LQ+63)/64, HD/32, BH), 32, 0, stream>>>(attn, vp, qp, outh);

  // 7) output projection (64x64 tiles)
  gemm_proj_kernel<<<dim3((MROWS+63)/64, DIM/64), 32, 0, stream>>>(outh, pjw, pjb, out);
}